// VectorQuantizer_49735721288439
// MI455X (gfx1250) — compile-verified
//
#include <hip/hip_runtime.h>
#include <hip/hip_bf16.h>

// VectorQuantizer for MI455X (gfx1250, wave32).
// Inputs : d_in[0] = z   (16*2048*256 f32)
//          d_in[1] = mask (16*2048 bool, 1 byte each)
//          d_in[2] = emb_weight W (1024*256 f32)
// Output : d_out = [ z_q (8388608 f32) | loss (1 f32) | indices-as-f32 (32768) ]
// Scratch: d_ws  = [ wnorm (1024 f32) | lossPart (2048 f32) | maskPart (2048 f32) ]

#define D_MODEL 256
#define N_BINS  1024
#define N_ROWS  32768           // B*T
#define ROWS_PER_BLOCK 16
#define N_BLOCKS (N_ROWS / ROWS_PER_BLOCK)   // 2048
#define LDS_STRIDE (D_MODEL + 2)             // 258 floats: de-conflict LDS banks

typedef float v2f __attribute__((ext_vector_type(2)));
typedef float v8f __attribute__((ext_vector_type(8)));

// ---------------------------------------------------------------------------
// Kernel 1: wnorm[j] = sum_d W[j][d]^2   (1024 blocks x 256 threads)
// ---------------------------------------------------------------------------
__global__ __launch_bounds__(256) void vq_wnorm_kernel(const float* __restrict__ W,
                                                       float* __restrict__ wnorm) {
    __shared__ float red[256];
    const int j = blockIdx.x;
    const int t = threadIdx.x;
    float v = W[(size_t)j * D_MODEL + t];
    red[t] = v * v;
    __syncthreads();
    #pragma unroll
    for (int s = 128; s > 0; s >>= 1) {
        if (t < s) red[t] += red[t + s];
        __syncthreads();
    }
    if (t == 0) wnorm[j] = red[0];
}

// ---------------------------------------------------------------------------
// Kernel 2: main fused GEMM + argmin + gather + partial loss.
// One workgroup (8 wave32) per 16 z-rows. The z-tile (16x256 f32 = 16KB) is
// staged into LDS once via GLOBAL_LOAD_ASYNC_TO_LDS_B128 (ASYNCcnt path) and
// re-read three times (znorm, WMMA A-fragments, loss). Wave w owns columns
// [w*128, w*128+128): 8 column tiles, f32 WMMA 16x16x4 over K=256.
// ---------------------------------------------------------------------------
__global__ __launch_bounds__(256) void vq_main_kernel(const float* __restrict__ z,
                                                      const unsigned char* __restrict__ mask,
                                                      const float* __restrict__ W,
                                                      const float* __restrict__ wnorm,
                                                      float* __restrict__ zq,
                                                      float* __restrict__ idx_out,
                                                      float* __restrict__ lossPart,
                                                      float* __restrict__ maskPart) {
    __shared__ float ztile[ROWS_PER_BLOCK][LDS_STRIDE];   // 16.5 KB
    __shared__ float znorm_s[ROWS_PER_BLOCK];
    __shared__ unsigned long long rowKey[ROWS_PER_BLOCK];
    __shared__ unsigned int idx_s[ROWS_PER_BLOCK];
    __shared__ float maskf_s[ROWS_PER_BLOCK];
    __shared__ float red[256];

    const int tid = threadIdx.x;
    const int rowBase = blockIdx.x * ROWS_PER_BLOCK;

    // --- Step 0: async-copy the 16x256 z-tile into LDS (padded rows) --------
    // 1024 chunks of 16B; 256 threads x 4 shots. ASYNCcnt-tracked DMA, no VGPR
    // round-trip. Each wave waits on its own ASYNCcnt, then block barrier.
    {
        const unsigned ldsBase = (unsigned)(uintptr_t)&ztile[0][0];
        #pragma unroll
        for (int i = 0; i < 4; ++i) {
            const int c   = tid + i * 256;   // chunk id 0..1023
            const int row = c >> 6;          // 0..15
            const int col = c & 63;          // 16B chunk within row
            unsigned lds = ldsBase + (unsigned)(row * (LDS_STRIDE * 4) + col * 16);
            unsigned long long ga =
                (unsigned long long)(uintptr_t)(z + (size_t)(rowBase + row) * D_MODEL + col * 4);
            asm volatile("global_load_async_to_lds_b128 %0, %1, off"
                         :: "v"(lds), "v"(ga) : "memory");
        }
        asm volatile("s_wait_asynccnt 0x0" ::: "memory");
    }
    if (tid < ROWS_PER_BLOCK) rowKey[tid] = ~0ull;
    __syncthreads();

    // --- Step A: row norms |z_m|^2 from LDS (fixed-order reduction) ---------
    {
        const int r = tid >> 4;          // 0..15
        const int seg = tid & 15;        // 16 elements each
        const float* p = &ztile[r][seg * 16];
        float s = 0.0f;
        #pragma unroll
        for (int i = 0; i < 16; ++i) s += p[i] * p[i];
        red[tid] = s;
    }
    __syncthreads();
    if (tid < ROWS_PER_BLOCK) {
        float s = 0.0f;
        #pragma unroll
        for (int i = 0; i < 16; ++i) s += red[tid * 16 + i];
        znorm_s[tid] = s;
    }
    __syncthreads();

    // --- Step B: S = Z_tile @ W^T via V_WMMA_F32_16X16X4_F32 ----------------
    const int lane = tid & 31;
    const int wv   = tid >> 5;           // wave id 0..7
    const int lm   = lane & 15;
    const int half = lane >> 4;          // 0: K{k,k+1} / 1: K{k+2,k+3}

    // A fragment from LDS: lane holds ztile[lm][k + 2*half .. +1]
    const float* Aptr = &ztile[lm][2 * half];
    // B fragment from global: W[wv*128 + t*16 + lm][k + 2*half .. +1]
    const float* Bptr = W + (size_t)(wv * 128 + lm) * D_MODEL + 2 * half;

    v8f acc[8];
    #pragma unroll
    for (int t = 0; t < 8; ++t) acc[t] = (v8f){0.f,0.f,0.f,0.f,0.f,0.f,0.f,0.f};

    #pragma unroll 4
    for (int k = 0; k < D_MODEL; k += 4) {
        v2f a = *(const v2f*)(Aptr + k);                     // ds_load_b64
        #pragma unroll
        for (int t = 0; t < 8; ++t) {
            v2f b = *(const v2f*)(Bptr + (size_t)t * 16 * D_MODEL + k);
            acc[t] = __builtin_amdgcn_wmma_f32_16x16x4_f32(
                /*neg_a=*/false, a, /*neg_b=*/false, b,
                /*c_mod=*/(short)0, acc[t],
                /*reuse_a=*/false, /*reuse_b=*/false);
        }
    }

    // --- Step C: per-lane argmin, then LDS ds_min_u64 across the block ------
    // C layout: VGPR v, lanes 0-15 -> row v, lanes 16-31 -> row v+8
    float wn[8];
    #pragma unroll
    for (int t = 0; t < 8; ++t) wn[t] = wnorm[wv * 128 + t * 16 + lm];

    #pragma unroll
    for (int v = 0; v < 8; ++v) {
        const int m = v + 8 * half;
        const float zn = znorm_s[m];
        unsigned long long best = ~0ull;
        #pragma unroll
        for (int t = 0; t < 8; ++t) {
            const int n = wv * 128 + t * 16 + lm;
            // mimic reference: d = (|z|^2 + |w|^2) - 2*(z.w)
            float d = (zn + wn[t]) - 2.0f * acc[t][v];
            unsigned long long key =
                ((unsigned long long)__float_as_uint(d) << 32) | (unsigned int)n;
            best = (key < best) ? key : best;
        }
        atomicMin(&rowKey[m], best);    // ds_min_u64 on LDS
    }
    __syncthreads();

    // --- Step D: extract indices, emit indices output -----------------------
    if (tid < ROWS_PER_BLOCK) {
        unsigned int idx = (unsigned int)(rowKey[tid] & 0xffffffffull);
        idx_s[tid] = idx;
        idx_out[rowBase + tid] = (float)idx;
        maskf_s[tid] = mask[rowBase + tid] ? 1.0f : 0.0f;
    }
    __syncthreads();

    // --- Step E: z_q gather + deterministic masked-SSE partial --------------
    float lsum = 0.0f;
    #pragma unroll 1
    for (int r = 0; r < ROWS_PER_BLOCK; ++r) {
        const unsigned int idx = idx_s[r];
        const size_t zi = (size_t)(rowBase + r) * D_MODEL + tid;
        float wval = W[(size_t)idx * D_MODEL + tid];
        float zval = ztile[r][tid];                          // from LDS
        zq[zi] = wval;
        float df = wval - zval;
        lsum += maskf_s[r] * df * df;
    }
    red[tid] = lsum;
    __syncthreads();
    #pragma unroll
    for (int s = 128; s > 0; s >>= 1) {
        if (tid < s) red[tid] += red[tid + s];
        __syncthreads();
    }
    if (tid == 0) {
        lossPart[blockIdx.x] = red[0];
        float mc = 0.0f;
        #pragma unroll
        for (int r = 0; r < ROWS_PER_BLOCK; ++r) mc += maskf_s[r];
        maskPart[blockIdx.x] = mc;
    }
}

// ---------------------------------------------------------------------------
// Kernel 3: deterministic final reduction -> loss scalar
// loss = (1 + beta) * SSE / max(sum(mask)*d_model, 1)
// ---------------------------------------------------------------------------
__global__ __launch_bounds__(256) void vq_loss_kernel(const float* __restrict__ lossPart,
                                                      const float* __restrict__ maskPart,
                                                      float* __restrict__ lossOut) {
    __shared__ float pl[256];
    __shared__ float pm[256];
    const int t = threadIdx.x;
    float sl = 0.0f, sm = 0.0f;
    #pragma unroll
    for (int i = 0; i < N_BLOCKS / 256; ++i) {   // 8 strided, fixed order
        sl += lossPart[t + i * 256];
        sm += maskPart[t + i * 256];
    }
    pl[t] = sl; pm[t] = sm;
    __syncthreads();
    #pragma unroll
    for (int s = 128; s > 0; s >>= 1) {
        if (t < s) { pl[t] += pl[t + s]; pm[t] += pm[t + s]; }
        __syncthreads();
    }
    if (t == 0) {
        float denom = fmaxf(pm[0] * (float)D_MODEL, 1.0f);
        lossOut[0] = 1.25f * pl[0] / denom;      // commit + 0.25*codebook (equal sums)
    }
}

// ---------------------------------------------------------------------------
extern "C" void kernel_launch(void* const* d_in, const int* in_sizes, int n_in,
                              void* d_out, int out_size, void* d_ws, size_t ws_size,
                              hipStream_t stream) {
    const float*         z    = (const float*)d_in[0];
    const unsigned char* mask = (const unsigned char*)d_in[1];   // jax bool = 1 byte
    const float*         W    = (const float*)d_in[2];

    float* out    = (float*)d_out;
    float* zq     = out;                               // 8,388,608 floats
    float* loss   = out + (size_t)N_ROWS * D_MODEL;    // 1 float
    float* idx_o  = loss + 1;                          // 32,768 floats

    float* ws       = (float*)d_ws;
    float* wnorm    = ws;                  // 1024
    float* lossPart = ws + N_BINS;         // 2048
    float* maskPart = lossPart + N_BLOCKS; // 2048

    vq_wnorm_kernel<<<N_BINS, 256, 0, stream>>>(W, wnorm);
    vq_main_kernel<<<N_BLOCKS, 256, 0, stream>>>(z, mask, W, wnorm,
                                                 zq, idx_o, lossPart, maskPart);
    vq_loss_kernel<<<1, 256, 0, stream>>>(lossPart, maskPart, loss);
}